// IzhikevichNeuron_52415780880474
// MI455X (gfx1250) — compile-verified
//
#include <hip/hip_runtime.h>
#include <stdint.h>

// Izhikevich scan: B=64, T=1024, D=512, float32 in/out.
// Memory-bound streaming recurrence; CDNA5 async-to-LDS double buffering.
// Roofline: 268 MB @ 23.3 TB/s => ~11.5 us floor; all effort goes to
// coalesced streaming + deep ASYNCcnt pipeline + NT stores.

#define B_SZ   64
#define T_SZ   1024
#define D_SZ   512
#define BLOCK  256          // threads per block (8 wave32s)
#define TILE   32           // timesteps staged per async batch (<= ASYNCcnt budget)
#define NTILES (T_SZ / TILE)

typedef __attribute__((address_space(3))) void lds_void;

__device__ __forceinline__ unsigned lds_addr_u32(void* p) {
    // generic -> LDS addrspace cast, then ptrtoint gives byte offset in LDS.
    return (unsigned)(uintptr_t)(lds_void*)p;
}

__device__ __forceinline__ void async_load_b32(unsigned lds_off, unsigned gl_off,
                                               const float* sbase) {
    // GVS form: reads global saddr (64-bit SGPR base) + vaddr (32-bit byte offset) into the LDS address in lds_off
    asm volatile("global_load_async_to_lds_b32 %0, %1, %2"
                 :: "v"(lds_off), "v"(gl_off), "s"(sbase)
                 : "memory");
}

__global__ __launch_bounds__(BLOCK) void izhikevich_scan_kernel(
    const float* __restrict__ I, float* __restrict__ O) {

    __shared__ float buf[2][TILE][BLOCK];   // 64 KB double buffer (320 KB/WGP avail)

    const int tid     = threadIdx.x;
    const int n0      = blockIdx.x * BLOCK;        // first neuron of this block
    const int b       = n0 / D_SZ;                 // batch row (uniform per block)
    const int dstart  = n0 % D_SZ;                 // d-chunk start (uniform)
    const size_t base = (size_t)b * T_SZ * D_SZ + (size_t)dstart;

    const float* sbase = I + base;                 // uniform -> SGPR pair
    float*       obase = O + base + tid;

    const unsigned lds0 = lds_addr_u32(&buf[0][0][0]);

    // Izhikevich state (per neuron, registers)
    float v = -65.0f;
    float u = 0.2f * -65.0f;

    // ---- prologue: stage 0 async loads (TILE outstanding per wave) ----
    {
        unsigned l = lds0 + (unsigned)tid * 4u;
        unsigned g = (unsigned)tid * 4u;
#pragma unroll
        for (int tt = 0; tt < TILE; ++tt) {
            async_load_b32(l, g, sbase);
            l += BLOCK * 4u;
            g += D_SZ * 4u;
        }
    }

    for (int tile = 0; tile < NTILES; ++tile) {
        const int stage = tile & 1;

        if (tile + 1 < NTILES) {
            // issue next stage, then wait until only those TILE remain in flight
            const int ns = stage ^ 1;
            unsigned l = lds0 + (unsigned)(ns * TILE * BLOCK + tid) * 4u;
            unsigned g = (unsigned)((tile + 1) * TILE * D_SZ + tid) * 4u;
#pragma unroll
            for (int tt = 0; tt < TILE; ++tt) {
                async_load_b32(l, g, sbase);
                l += BLOCK * 4u;
                g += D_SZ * 4u;
            }
            asm volatile("s_wait_asynccnt %0" :: "n"(TILE) : "memory");
        } else {
            asm volatile("s_wait_asynccnt 0" ::: "memory");
        }

        float* op = obase + (size_t)tile * TILE * D_SZ;
#pragma unroll
        for (int tt = 0; tt < TILE; ++tt) {
            const float it = buf[stage][tt][tid];          // ds_load (pairs into 2addr)
            // dv = (0.04v + 5)v + 140 - u + i   -> 2 FMA + add + sub
            float a  = __fmaf_rn(0.04f, v, 5.0f);
            float bq = __fmaf_rn(a, v, 140.0f);
            float dv = (bq + it) - u;
            v = __fmaf_rn(0.2f, dv, v);                    // v += DT*dv
            // u += DT*A*(B*v - u) = u + 0.004*(0.2*v - u) ; -u is a free src modifier
            float du = __fmaf_rn(0.2f, v, -u);
            u = __fmaf_rn(0.004f, du, u);
            const bool spike = (v >= 30.0f);
            const float s = spike ? 1.0f : 0.0f;
            v = spike ? -65.0f : v;
            u = spike ? (u + 6.0f) : u;
            __builtin_nontemporal_store(s, op);            // streaming store, TH=NT
            op += D_SZ;                                    // folds into 24-bit ioffset
        }
    }
}

extern "C" void kernel_launch(void* const* d_in, const int* in_sizes, int n_in,
                              void* d_out, int out_size, void* d_ws, size_t ws_size,
                              hipStream_t stream) {
    (void)in_sizes; (void)n_in; (void)d_ws; (void)ws_size; (void)out_size;
    const float* I = (const float*)d_in[0];
    float*       O = (float*)d_out;
    const int nblocks = (B_SZ * D_SZ) / BLOCK;   // 128 blocks x 256 threads = 32768 neurons
    izhikevich_scan_kernel<<<nblocks, BLOCK, 0, stream>>>(I, O);
}